// ZXNet_65317862637645
// MI455X (gfx1250) — compile-verified
//
#include <hip/hip_runtime.h>
#include <hip/hip_bf16.h>
#include <math.h>

typedef __attribute__((ext_vector_type(2))) float v2f;
typedef __attribute__((ext_vector_type(8))) float v8f;

#define LDS_STRIDE 72   // 64 + 8: rows K and K+2 hit disjoint LDS bank sets

// ---------------------------------------------------------------------------
// C[n_rows x 64] = X[n_rows x K] @ W[K x 64]   (row-major, K multiple of 4, K<=192)
// grid.x = ceil(n_rows/32), block = 256 (8 wave32s). wave w: mtile=w>>2, ntile=w&3.
// fp32 tensor path: V_WMMA_F32_16X16X4_F32 per the CDNA5 ISA layouts:
//   A 16x4 : lanes 0-15 hold (K=0,K=1), lanes 16-31 hold (K=2,K=3) for row M=lane%16
//   B 4x16 : lanes 0-15 hold rows K=0/K=1, lanes 16-31 rows K=2/K=3, col N=lane%16
//   C/D    : VGPR r -> (M=r, N=lane) for lanes 0-15, (M=r+8, N=lane-16) for 16-31
// ---------------------------------------------------------------------------
__global__ __launch_bounds__(256)
void gcn_gemm_wmma(const float* __restrict__ X, const float* __restrict__ W,
                   float* __restrict__ C, int n_rows, int K) {
  __shared__ float sW[192 * LDS_STRIDE];
  for (int i = threadIdx.x; i < K * 64; i += 256)
    sW[(i >> 6) * LDS_STRIDE + (i & 63)] = W[i];
  __syncthreads();

  const int wave   = threadIdx.x >> 5;
  const int lane   = threadIdx.x & 31;
  const int lanelo = lane & 15;
  const int hi     = lane >> 4;        // 0: K+0/1, 1: K+2/3
  const int mtile  = wave >> 2;
  const int ntile  = wave & 3;
  const int row0   = blockIdx.x * 32 + mtile * 16;

  int arow = row0 + lanelo;
  if (arow >= n_rows) arow = n_rows - 1;        // clamp reads; EXEC stays all-1 for WMMA
  const float* __restrict__ xrow = X + (size_t)arow * K;
  const int col = ntile * 16 + lanelo;

  v8f acc = {0.f, 0.f, 0.f, 0.f, 0.f, 0.f, 0.f, 0.f};
  for (int k = 0; k < K; k += 4) {
    const int ka = k + hi * 2;
    v2f a, b;
    a.x = xrow[ka];
    a.y = xrow[ka + 1];
    b.x = sW[ka * LDS_STRIDE + col];
    b.y = sW[(ka + 1) * LDS_STRIDE + col];
    acc = __builtin_amdgcn_wmma_f32_16x16x4_f32(false, a, false, b,
                                                (short)0, acc, false, false);
  }

  const int mbase = row0 + hi * 8;
#pragma unroll
  for (int r = 0; r < 8; ++r) {
    const int m = mbase + r;
    if (m < n_rows) C[(size_t)m * 64 + ntile * 16 + lanelo] = acc[r];
  }
}

// deg[dst[e]] += 1  (deg buffer pre-zeroed)
__global__ void gcn_degree(const int* __restrict__ dst, float* __restrict__ deg, int nedges) {
  int e = blockIdx.x * blockDim.x + threadIdx.x;
  if (e < nedges) atomicAdd(&deg[dst[e]], 1.0f);
}

// in place: d[i] = rsqrt(d[i] + 1)
__global__ void gcn_rsqrt(float* __restrict__ d, int n) {
  int i = blockIdx.x * blockDim.x + threadIdx.x;
  if (i < n) d[i] = rsqrtf(d[i] + 1.0f);
}

// One wave32 per edge; lane handles feature pair (2*lane, 2*lane+1).
// agg[dst] += dinv[src]*dinv[dst] * xw[src]   (64 features, L2-resident atomics)
__global__ __launch_bounds__(256)
void gcn_edge_scatter(const int* __restrict__ src, const int* __restrict__ dst,
                      const float* __restrict__ dinv, const float* __restrict__ xw,
                      float* __restrict__ agg, int nedges) {
  const int wid  = (int)(((size_t)blockIdx.x * blockDim.x + threadIdx.x) >> 5);
  const int lane = threadIdx.x & 31;
  if (wid >= nedges) return;
  const int s = src[wid];
  const int d = dst[wid];
  const float w = dinv[s] * dinv[d];
  const float2 v = *reinterpret_cast<const float2*>(xw + (size_t)s * 64 + lane * 2);
  float* out = agg + (size_t)d * 64 + lane * 2;
  atomicAdd(out,     w * v.x);
  atomicAdd(out + 1, w * v.y);
}

// agg := relu(agg + dinv^2 * xw + bias)   (thread per node x feature-pair)
__global__ void gcn_finalize(float* __restrict__ agg, const float* __restrict__ xw,
                             const float* __restrict__ dinv, const float* __restrict__ bias,
                             int n) {
  int idx = blockIdx.x * blockDim.x + threadIdx.x;
  if (idx >= n * 32) return;
  const int node = idx >> 5;
  const int f2   = (idx & 31) * 2;
  const float di = dinv[node];
  const float s  = di * di;
  const size_t o = (size_t)node * 64 + f2;
  float h0 = agg[o]     + s * xw[o]     + bias[f2];
  float h1 = agg[o + 1] + s * xw[o + 1] + bias[f2 + 1];
  agg[o]     = h0 > 0.f ? h0 : 0.f;
  agg[o + 1] = h1 > 0.f ? h1 : 0.f;
}

// One wave per node: sums[batch[i]] += h[i]; cnt[batch[i]] += 1
__global__ __launch_bounds__(256)
void gcn_pool_scatter(const float* __restrict__ h, const int* __restrict__ batch,
                      float* __restrict__ sums, float* __restrict__ cnt, int n) {
  const int wid  = (int)(((size_t)blockIdx.x * blockDim.x + threadIdx.x) >> 5);
  const int lane = threadIdx.x & 31;
  if (wid >= n) return;
  const int g = batch[wid];
  const float2 v = *reinterpret_cast<const float2*>(h + (size_t)wid * 64 + lane * 2);
  atomicAdd(&sums[(size_t)g * 64 + lane * 2],     v.x);
  atomicAdd(&sums[(size_t)g * 64 + lane * 2 + 1], v.y);
  if (lane == 0) atomicAdd(&cnt[g], 1.0f);
}

__global__ void gcn_pool_div(const float* __restrict__ sums, const float* __restrict__ cnt,
                             float* __restrict__ emb, int Gn) {
  int idx = blockIdx.x * blockDim.x + threadIdx.x;
  if (idx >= Gn * 64) return;
  const int g = idx >> 6;
  const float c = cnt[g];
  emb[idx] = sums[idx] / (c > 1.0f ? c : 1.0f);
}

// combined[g] = [emb1 | emb2 | |emb1-emb2|]   (192 wide)
__global__ void gcn_combine(const float* __restrict__ e1, const float* __restrict__ e2,
                            float* __restrict__ comb, int Gn) {
  int idx = blockIdx.x * blockDim.x + threadIdx.x;
  if (idx >= Gn * 64) return;
  const int g = idx >> 6;
  const int f = idx & 63;
  const float a = e1[idx];
  const float b = e2[idx];
  float* row = comb + (size_t)g * 192;
  row[f]        = a;
  row[64 + f]   = b;
  row[128 + f]  = fabsf(a - b);
}

// out[g] = sigmoid( sum_f relu(hid[g,f]+fc1_b[f]) * fc2_W[f] + fc2_b )  — wave per graph
__global__ __launch_bounds__(256)
void gcn_head_final(const float* __restrict__ hid, const float* __restrict__ fc1b,
                    const float* __restrict__ fc2W, const float* __restrict__ fc2b,
                    float* __restrict__ out, int Gn) {
  const int wid  = (int)(((size_t)blockIdx.x * blockDim.x + threadIdx.x) >> 5);
  const int lane = threadIdx.x & 31;
  if (wid >= Gn) return;
  float p = 0.f;
#pragma unroll
  for (int j = 0; j < 2; ++j) {
    const int f = lane * 2 + j;
    float v = hid[(size_t)wid * 64 + f] + fc1b[f];
    v = v > 0.f ? v : 0.f;
    p += v * fc2W[f];
  }
#pragma unroll
  for (int o = 16; o > 0; o >>= 1) p += __shfl_xor(p, o, 32);
  if (lane == 0) out[wid] = 1.0f / (1.0f + __expf(-(p + fc2b[0])));
}

// ---------------------------------------------------------------------------
static void run_branch(const float* x, const int* src, const int* dst, const int* batch,
                       const float* W1, const float* b1, const float* W2, const float* b2,
                       float* A, float* B, float* dinv, float* sums, float* cnt, float* emb,
                       int N, int E, int F_IN, int G, hipStream_t stream) {
  const size_t featBytes = (size_t)N * 64 * sizeof(float);

  // degrees -> dinv
  hipMemsetAsync(dinv, 0, (size_t)N * sizeof(float), stream);
  gcn_degree<<<(E + 255) / 256, 256, 0, stream>>>(dst, dinv, E);
  gcn_rsqrt<<<(N + 255) / 256, 256, 0, stream>>>(dinv, N);

  // layer 1: A = x @ W1 ; B = scatter(norm * A[src]) ; B = relu(B + dinv^2*A + b1)
  gcn_gemm_wmma<<<(N + 31) / 32, 256, 0, stream>>>(x, W1, A, N, F_IN);
  hipMemsetAsync(B, 0, featBytes, stream);
  gcn_edge_scatter<<<(E + 7) / 8, 256, 0, stream>>>(src, dst, dinv, A, B, E);
  gcn_finalize<<<((size_t)N * 32 + 255) / 256, 256, 0, stream>>>(B, A, dinv, b1, N);

  // layer 2: A = B @ W2 ; B = scatter(norm * A[src]) ; B = relu(B + dinv^2*A + b2)
  gcn_gemm_wmma<<<(N + 31) / 32, 256, 0, stream>>>(B, W2, A, N, 64);
  hipMemsetAsync(B, 0, featBytes, stream);
  gcn_edge_scatter<<<(E + 7) / 8, 256, 0, stream>>>(src, dst, dinv, A, B, E);
  gcn_finalize<<<((size_t)N * 32 + 255) / 256, 256, 0, stream>>>(B, A, dinv, b2, N);

  // mean pool
  hipMemsetAsync(sums, 0, (size_t)G * 64 * sizeof(float), stream);
  hipMemsetAsync(cnt, 0, (size_t)G * sizeof(float), stream);
  gcn_pool_scatter<<<(N + 7) / 8, 256, 0, stream>>>(B, batch, sums, cnt, N);
  gcn_pool_div<<<((size_t)G * 64 + 255) / 256, 256, 0, stream>>>(sums, cnt, emb, G);
}

extern "C" void kernel_launch(void* const* d_in, const int* in_sizes, int n_in,
                              void* d_out, int out_size, void* d_ws, size_t ws_size,
                              hipStream_t stream) {
  const float* x1     = (const float*)d_in[0];
  const int*   ei1    = (const int*)d_in[1];
  const int*   batch1 = (const int*)d_in[2];
  const float* x2     = (const float*)d_in[3];
  const int*   ei2    = (const int*)d_in[4];
  const int*   batch2 = (const int*)d_in[5];
  const float* W1     = (const float*)d_in[6];
  const float* b1     = (const float*)d_in[7];
  const float* W2     = (const float*)d_in[8];
  const float* b2     = (const float*)d_in[9];
  const float* fc1W   = (const float*)d_in[10];
  const float* fc1b   = (const float*)d_in[11];
  const float* fc2W   = (const float*)d_in[12];
  const float* fc2b   = (const float*)d_in[13];
  float* out = (float*)d_out;

  const int H    = in_sizes[7];           // 64
  const int F_IN = in_sizes[6] / H;       // 128
  const int N    = in_sizes[0] / F_IN;    // 100000
  const int E    = in_sizes[1] / 2;       // 3.2M
  const int G    = out_size;              // 1024
  (void)n_in; (void)ws_size; (void)H;

  // workspace carve-up (floats)
  float* ws   = (float*)d_ws;
  float* A    = ws;                          // N*64  (xw ping)
  float* B    = A    + (size_t)N * 64;       // N*64  (agg/h pong)
  float* dinv = B    + (size_t)N * 64;       // N
  float* sums = dinv + (size_t)N;            // G*64
  float* cnt  = sums + (size_t)G * 64;       // G
  float* emb1 = cnt  + (size_t)G;            // G*64
  float* emb2 = emb1 + (size_t)G * 64;       // G*64
  float* comb = emb2 + (size_t)G * 64;       // G*192
  float* hid  = comb + (size_t)G * 192;      // G*64

  const int* src1 = ei1;           const int* dst1 = ei1 + E;
  const int* src2 = ei2;           const int* dst2 = ei2 + E;

  run_branch(x1, src1, dst1, batch1, W1, b1, W2, b2,
             A, B, dinv, sums, cnt, emb1, N, E, F_IN, G, stream);
  run_branch(x2, src2, dst2, batch2, W1, b1, W2, b2,
             A, B, dinv, sums, cnt, emb2, N, E, F_IN, G, stream);

  // head: comb[G x 192] -> hid[G x 64] (WMMA) -> sigmoid scalar
  gcn_combine<<<((size_t)G * 64 + 255) / 256, 256, 0, stream>>>(emb1, emb2, comb, G);
  gcn_gemm_wmma<<<(G + 31) / 32, 256, 0, stream>>>(comb, fc1W, hid, G, 192);
  gcn_head_final<<<(G + 7) / 8, 256, 0, stream>>>(hid, fc1b, fc2W, fc2b, out, G);
}